// Seq2Seq_19765439496259
// MI455X (gfx1250) — compile-verified
//
#include <hip/hip_runtime.h>
#include <hip/hip_bf16.h>
#include <cstdint>

// ---------------------------------------------------------------------------
// Seq2Seq LSTM (E=H=1024, L=2, S=T=2048, V=50257), all f32.
//  - Gate pre-GEMMs and the final vocab projection use V_WMMA_F32_16X16X4_F32.
//  - The sequential LSTM scans run as a persistent 256-WG kernel with Whh
//    pinned in LDS (16 gate-rows/WG, 320KB WGP LDS) and a grid-wide atomic
//    barrier per timestep.
// ---------------------------------------------------------------------------

typedef __attribute__((ext_vector_type(2))) float v2f;
typedef __attribute__((ext_vector_type(8))) float v8f;

#define KDIM 1024          // inner dim for every GEMM (E == H == 1024)
#define GATES 4096         // 4*H
#define SEQ 2048
#define VOCAB 50257
#define SCAN_WGS 256
#define WSP 1040           // padded LDS row stride for Whh (1024 + 16)

// ----------------------------- small kernels -------------------------------

__global__ __launch_bounds__(256) void embed_enc_kernel(
    const int* __restrict__ tok, const float* __restrict__ emb,
    float* __restrict__ out) {
  int s = blockIdx.x;
  int v = tok[s];
  const float4* src = (const float4*)(emb + (size_t)v * KDIM);
  float4* dst = (float4*)(out + (size_t)s * KDIM);
  dst[threadIdx.x] = src[threadIdx.x];   // 256 threads x float4 = 1024 floats
}

__global__ __launch_bounds__(256) void embed_dec_kernel(
    const int* __restrict__ outsTok, const int* __restrict__ sosPtr,
    const float* __restrict__ emb, float* __restrict__ out) {
  int t = blockIdx.x;
  int v = (t == 0) ? sosPtr[0] : outsTok[t - 1];
  const float4* src = (const float4*)(emb + (size_t)v * KDIM);
  float4* dst = (float4*)(out + (size_t)t * KDIM);
  dst[threadIdx.x] = src[threadIdx.x];
}

__global__ __launch_bounds__(256) void add_bias_kernel(
    const float* __restrict__ a, const float* __restrict__ b,
    float* __restrict__ o, int n) {
  int i = blockIdx.x * 256 + threadIdx.x;
  if (i < n) o[i] = a[i] + b[i];
}

// --------------------------- WMMA f32 GEMM ---------------------------------
// C[M, ldc] = A[M,1024] * B^T (B stored [N,1024] row-major) + bias[N]
// 256 threads = 8 waves; 64x64 tile/WG; each wave does a 16x32 sub-tile.
// Fragment layouts per CDNA5 ISA 7.12.2:
//   A 16x4 f32 : lanes 0-15 -> (K0,K1), lanes 16-31 -> (K2,K3), M = lane&15
//   B  4x16    : VGPR i holds rows K=i (lanes 0-15) and K=i+2 (lanes 16-31)
//   -> with B^T stored [n][k] in LDS, A and B frags share one gather pattern.
//   C 16x16    : VGPR r: lane<16 -> (M=r, N=lane); lane>=16 -> (M=r+8, N=lane-16)

__global__ __launch_bounds__(256) void wmma_gemm_btn_kernel(
    const float* __restrict__ A, const float* __restrict__ B,
    const float* __restrict__ bias, float* __restrict__ C,
    int Ntot, int ldc) {
  __shared__ float As[64][20];   // 16 K + pad to 20 (keeps float4 align, no conflicts)
  __shared__ float Bs[64][20];

  const int tid  = threadIdx.x;
  const int wave = tid >> 5;
  const int lane = tid & 31;
  const int half = lane >> 4;    // 0: lanes 0-15, 1: lanes 16-31
  const int lm   = lane & 15;

  const int m0 = blockIdx.y * 64;
  const int n0 = blockIdx.x * 64;
  const int mo = (wave >> 1) * 16;   // wave M offset within tile
  const int no = (wave & 1) * 32;    // wave N offset within tile

  const int lr = tid >> 2;           // load row 0..63
  const int lk = (tid & 3) * 4;      // load k offset 0,4,8,12

  v8f acc0 = {};
  v8f acc1 = {};

  for (int kk = 0; kk < KDIM; kk += 16) {
    float4 av = *(const float4*)(A + (size_t)(m0 + lr) * KDIM + kk + lk);
    int brow = n0 + lr;
    float4 bv = make_float4(0.f, 0.f, 0.f, 0.f);
    if (brow < Ntot) bv = *(const float4*)(B + (size_t)brow * KDIM + kk + lk);
    if (kk + 16 < KDIM) {  // software prefetch of next K chunk (global_prefetch_b8)
      __builtin_prefetch(A + (size_t)(m0 + lr) * KDIM + kk + 16 + lk, 0, 1);
      if (brow < Ntot)
        __builtin_prefetch(B + (size_t)brow * KDIM + kk + 16 + lk, 0, 1);
    }
    __syncthreads();   // protect previous iteration's LDS reads
    *(float4*)(&As[lr][lk]) = av;
    *(float4*)(&Bs[lr][lk]) = bv;
    __syncthreads();

#pragma unroll
    for (int k4 = 0; k4 < 16; k4 += 4) {
      const int kc = k4 + 2 * half;
      v2f a, b0, b1;
      a.x  = As[mo + lm][kc];       a.y  = As[mo + lm][kc + 1];
      b0.x = Bs[no + lm][kc];       b0.y = Bs[no + lm][kc + 1];
      b1.x = Bs[no + 16 + lm][kc];  b1.y = Bs[no + 16 + lm][kc + 1];
      acc0 = __builtin_amdgcn_wmma_f32_16x16x4_f32(false, a, false, b0,
                                                   (short)0, acc0, false, false);
      acc1 = __builtin_amdgcn_wmma_f32_16x16x4_f32(false, a, false, b1,
                                                   (short)0, acc1, false, false);
    }
  }

#pragma unroll
  for (int r = 0; r < 8; ++r) {
    const int m  = m0 + mo + r + 8 * half;
    const int n  = n0 + no + lm;
    const int n2 = n + 16;
    if (n  < Ntot) C[(size_t)m * ldc + n]  = acc0[r] + bias[n];
    if (n2 < Ntot) C[(size_t)m * ldc + n2] = acc1[r] + bias[n2];
  }
}

// --------------------------- persistent LSTM scan --------------------------

__device__ inline void grid_barrier(unsigned* bar, unsigned nwg) {
  __threadfence();
  __syncthreads();
  if (threadIdx.x == 0) {
    unsigned* cnt = bar;
    unsigned* gen = bar + 1;
    unsigned g = __hip_atomic_load(gen, __ATOMIC_ACQUIRE, __HIP_MEMORY_SCOPE_AGENT);
    unsigned v = __hip_atomic_fetch_add(cnt, 1u, __ATOMIC_ACQ_REL,
                                        __HIP_MEMORY_SCOPE_AGENT);
    if (v == nwg - 1) {
      __hip_atomic_store(cnt, 0u, __ATOMIC_RELAXED, __HIP_MEMORY_SCOPE_AGENT);
      __hip_atomic_fetch_add(gen, 1u, __ATOMIC_RELEASE, __HIP_MEMORY_SCOPE_AGENT);
    } else {
      while (__hip_atomic_load(gen, __ATOMIC_ACQUIRE,
                               __HIP_MEMORY_SCOPE_AGENT) == g) {
        __builtin_amdgcn_s_sleep(1);
      }
    }
  }
  __syncthreads();
  __threadfence();
}

// 256 WGs x 256 threads. WG w owns h-indices n in [4w, 4w+4) and the 16 gate
// rows {g*1024 + n : g in 0..3}. Whh rows live in LDS for the whole scan; c in
// registers of threads 0..3; h exchanged via a 4KB global buffer each step.
// gatesX[t, 4096] already holds Wih@x_t + bih + bhh.
__global__ __launch_bounds__(256) void lstm_scan_kernel(
    const float* __restrict__ gatesX, const float* __restrict__ Whh,
    const float* __restrict__ hInit, const float* __restrict__ cInit,
    float* __restrict__ outSeq, float* __restrict__ hFin,
    float* __restrict__ cFin, float* __restrict__ hBuf,
    unsigned* bar, int T) {
  extern __shared__ float smem[];
  float* Ws   = smem;                 // 16 rows x WSP
  float* hSh  = smem + 16 * WSP;      // 1024
  float* gsum = hSh + KDIM;           // 16

  const int tid    = threadIdx.x;
  const int nBase  = blockIdx.x * 4;
  const int rowL   = tid >> 4;        // 0..15 : gate-row within WG
  const int lane16 = tid & 15;

  // Pin this WG's 16 Whh rows (row r -> global row (r>>2)*1024 + nBase + (r&3))
  for (int idx = tid * 4; idx < 16 * KDIM; idx += 256 * 4) {
    int r = idx >> 10;
    int k = idx & (KDIM - 1);
    int j = ((r >> 2) << 10) + nBase + (r & 3);
    *(float4*)(&Ws[r * WSP + k]) = *(const float4*)(&Whh[(size_t)j * KDIM + k]);
  }

  float c = 0.f, hOut = 0.f;
  if (tid < 4) {
    c = cInit[nBase + tid];
    hBuf[nBase + tid] = hInit[nBase + tid];
  }
  grid_barrier(bar, gridDim.x);

  for (int t = 0; t < T; ++t) {
    for (int k = tid; k < KDIM; k += 256) hSh[k] = hBuf[k];
    __syncthreads();

    const float* wr = &Ws[rowL * WSP];
    float acc = 0.f;
#pragma unroll 8
    for (int i = 0; i < 64; ++i) {
      int k = (i << 4) + lane16;
      acc = fmaf(wr[k], hSh[k], acc);
    }
#pragma unroll
    for (int off = 8; off; off >>= 1) acc += __shfl_xor(acc, off, 16);
    if (lane16 == 0) gsum[rowL] = acc;
    __syncthreads();

    if (tid < 4) {
      const int n = nBase + tid;
      const float* gx = gatesX + (size_t)t * GATES;
      float gi = gsum[0 * 4 + tid] + gx[n];
      float gf = gsum[1 * 4 + tid] + gx[1024 + n];
      float gg = gsum[2 * 4 + tid] + gx[2048 + n];
      float go = gsum[3 * 4 + tid] + gx[3072 + n];
      float si = 1.f / (1.f + __expf(-gi));
      float sf = 1.f / (1.f + __expf(-gf));
      float so = 1.f / (1.f + __expf(-go));
      c    = sf * c + si * tanhf(gg);
      hOut = so * tanhf(c);
      hBuf[n] = hOut;
      outSeq[(size_t)t * KDIM + n] = hOut;
    }
    grid_barrier(bar, gridDim.x);
  }

  if (tid < 4) {
    hFin[nBase + tid] = hOut;
    cFin[nBase + tid] = c;
  }
}

// ------------------------------- launcher ----------------------------------

extern "C" void kernel_launch(void* const* d_in, const int* in_sizes, int n_in,
                              void* d_out, int out_size, void* d_ws,
                              size_t ws_size, hipStream_t stream) {
  (void)in_sizes; (void)n_in; (void)out_size; (void)ws_size;
  const int*   inputs  = (const int*)d_in[0];
  const int*   outputs = (const int*)d_in[1];
  const int*   sos     = (const int*)d_in[2];
  const float* enc_emb = (const float*)d_in[3];
  const float* dec_emb = (const float*)d_in[4];
  const float* enc_Wih = (const float*)d_in[5];
  const float* enc_Whh = (const float*)d_in[6];
  const float* enc_bih = (const float*)d_in[7];
  const float* enc_bhh = (const float*)d_in[8];
  const float* dec_Wih = (const float*)d_in[9];
  const float* dec_Whh = (const float*)d_in[10];
  const float* dec_bih = (const float*)d_in[11];
  const float* dec_bhh = (const float*)d_in[12];
  const float* lin_W   = (const float*)d_in[13];
  const float* lin_b   = (const float*)d_in[14];
  float* out = (float*)d_out;
  float* ws  = (float*)d_ws;

  // Workspace layout (floats); ~50.5 MB total.
  const size_t SEQF = (size_t)SEQ * KDIM;       // 2,097,152
  float* xA    = ws;
  float* xB    = xA + SEQF;
  float* gates = xB + SEQF;                     // SEQ * 4096
  float* encB  = gates + (size_t)SEQ * GATES;   // 2*4096
  float* decB  = encB + 2 * GATES;
  float* hFin  = decB + 2 * GATES;              // [2][1024]
  float* cFin  = hFin + 2 * KDIM;
  float* hTmp  = cFin + 2 * KDIM;               // [1024] (decoder finals, unused)
  float* cTmp  = hTmp + KDIM;
  float* hBuf  = cTmp + KDIM;                   // [1024]
  float* zbuf  = hBuf + KDIM;                   // [1024] zeros
  unsigned* bar = (unsigned*)(zbuf + KDIM);     // [16]

  const size_t scanLds = (size_t)(16 * WSP + KDIM + 16) * sizeof(float); // ~70.7 KB
  hipFuncSetAttribute((const void*)lstm_scan_kernel,
                      hipFuncAttributeMaxDynamicSharedMemorySize, (int)scanLds);

  hipMemsetAsync(bar, 0, 16 * sizeof(unsigned), stream);
  hipMemsetAsync(zbuf, 0, KDIM * sizeof(float), stream);

  add_bias_kernel<<<(2 * GATES + 255) / 256, 256, 0, stream>>>(
      enc_bih, enc_bhh, encB, 2 * GATES);
  add_bias_kernel<<<(2 * GATES + 255) / 256, 256, 0, stream>>>(
      dec_bih, dec_bhh, decB, 2 * GATES);

  const dim3 gGates(GATES / 64, SEQ / 64);                  // 64 x 32
  const dim3 gLogit((VOCAB + 63) / 64, SEQ / 64);           // 786 x 32
  const size_t WL = (size_t)GATES * KDIM;                   // per-layer weight stride

  // ---- Encoder ----
  embed_enc_kernel<<<SEQ, 256, 0, stream>>>(inputs, enc_emb, xA);

  wmma_gemm_btn_kernel<<<gGates, 256, 0, stream>>>(xA, enc_Wih, encB, gates,
                                                   GATES, GATES);
  lstm_scan_kernel<<<SCAN_WGS, 256, scanLds, stream>>>(
      gates, enc_Whh, zbuf, zbuf, xB, hFin, cFin, hBuf, bar, SEQ);

  wmma_gemm_btn_kernel<<<gGates, 256, 0, stream>>>(xB, enc_Wih + WL,
                                                   encB + GATES, gates, GATES,
                                                   GATES);
  lstm_scan_kernel<<<SCAN_WGS, 256, scanLds, stream>>>(
      gates, enc_Whh + WL, zbuf, zbuf, xA, hFin + KDIM, cFin + KDIM, hBuf, bar,
      SEQ);

  // ---- Decoder (teacher forced: [sos, outputs[:-1]]) ----
  embed_dec_kernel<<<SEQ, 256, 0, stream>>>(outputs, sos, dec_emb, xA);

  wmma_gemm_btn_kernel<<<gGates, 256, 0, stream>>>(xA, dec_Wih, decB, gates,
                                                   GATES, GATES);
  lstm_scan_kernel<<<SCAN_WGS, 256, scanLds, stream>>>(
      gates, dec_Whh, hFin, cFin, xB, hTmp, cTmp, hBuf, bar, SEQ);

  wmma_gemm_btn_kernel<<<gGates, 256, 0, stream>>>(xB, dec_Wih + WL,
                                                   decB + GATES, gates, GATES,
                                                   GATES);
  lstm_scan_kernel<<<SCAN_WGS, 256, scanLds, stream>>>(
      gates, dec_Whh + WL, hFin + KDIM, cFin + KDIM, xA, hTmp, cTmp, hBuf, bar,
      SEQ);

  // ---- Vocab projection ----
  wmma_gemm_btn_kernel<<<gLogit, 256, 0, stream>>>(xA, lin_W, lin_b, out,
                                                   VOCAB, VOCAB);
}